// DecoderLSTM_19988777796313
// MI455X (gfx1250) — compile-verified
//
#include <hip/hip_runtime.h>
#include <math.h>

// Sizes from the reference: B=32, S=128, H=1024, L=2, V=32000
#define HD 1024
#define BD 32
#define SD 128
#define VD 32000

typedef float v2f __attribute__((ext_vector_type(2)));
typedef float v8f __attribute__((ext_vector_type(8)));

__device__ __forceinline__ float sigf(float x) { return 1.0f / (1.0f + __expf(-x)); }

// One LSTM time-step for one layer, fully fused, with 4-way K-split per tile:
//   z = x @ W_ih^T + h_prev @ W_hh^T + b_ih + b_hh       (f32 WMMA 16x16x4)
//   c = sig(f)*c + sig(i)*tanh(g);  h = sig(o)*tanh(c)   (fused epilogue)
// Tiles: 2 M-tiles (B=32) x 64 h-tiles (H=1024) = 128 tiles. Each tile is
// computed by 4 waves, each covering K/4 = 256; partials are combined through
// LDS (fixed order -> deterministic) and the kq==0 wave runs the cell update.
// 512 waves = 64 blocks x 256 threads; serial WMMA chain per wave = 512.
__global__ __launch_bounds__(256)
void lstm_step_kernel(const float* __restrict__ emb,
                      const int*   __restrict__ tok, int tok_stride,
                      const float* __restrict__ xin,
                      const float* __restrict__ hprev,
                      float*       __restrict__ cstate,
                      const float* __restrict__ wih,
                      const float* __restrict__ whh,
                      const float* __restrict__ bih,
                      const float* __restrict__ bhh,
                      float*       __restrict__ hout,
                      float*       __restrict__ seqout) // already offset by t*HD; null for layer 0
{
    __shared__ float red[2][3][4][8][32];   // [tileInBlock][kq-1][gate][r][lane], 24 KB

    const int lane   = threadIdx.x & 31;
    const int wib    = threadIdx.x >> 5;        // wave in block: 0..7
    const int tib    = wib >> 2;                // tile in block: 0..1
    const int kq     = wib & 3;                 // K quarter: 0..3
    const int tile   = blockIdx.x * 2 + tib;    // 0..127
    const int h_tile = tile & 63;
    const int m_tile = tile >> 6;               // 0..1
    const int lane15 = lane & 15;
    const int hiHalf = (lane >> 4) & 1;         // lanes 16..31 hold K = k+2,k+3
    const int koff   = hiHalf * 2;

    const int arow = m_tile * 16 + lane15;      // batch row this lane supplies to A
    const int hcol = h_tile * 16 + lane15;      // hidden column this lane supplies to B
    const int k0   = kq * (HD / 4);
    const int k1   = k0 + (HD / 4);

    // A operand base pointers (x input and recurrent h input)
    const float* ax;
    if (tok) {                                  // layer 0: embedded token
        const int tb = tok[(size_t)arow * tok_stride];
        ax = emb + (size_t)tb * HD + koff;
    } else {                                    // layer 1: layer-0 output this step
        ax = xin + (size_t)arow * HD + koff;
    }
    const float* ah = hprev + (size_t)arow * HD + koff;

    // B operand base pointers, one per gate (rows of W are contiguous along K)
    const float* bx[4];
    const float* bh[4];
#pragma unroll
    for (int g = 0; g < 4; ++g) {
        const int n = g * HD + hcol;            // row in [4H, H] weight matrix
        bx[g] = wih + (size_t)n * HD + koff;
        bh[g] = whh + (size_t)n * HD + koff;
    }

    // kq==0 wave seeds its accumulators with the (N-broadcast) bias
    v8f acc[4];
#pragma unroll
    for (int g = 0; g < 4; ++g) {
        float bz = 0.0f;
        if (kq == 0) {
            const int n = g * HD + hcol;
            bz = bih[n] + bhh[n];
        }
        acc[g] = (v8f){bz, bz, bz, bz, bz, bz, bz, bz};
    }

    // K loop over this wave's quarter: 64 iterations, 8 f32 WMMAs each
#pragma unroll 2
    for (int k = k0; k < k1; k += 4) {
        const v2f a0 = *(const v2f*)(ax + k);
        const v2f a1 = *(const v2f*)(ah + k);
#pragma unroll
        for (int g = 0; g < 4; ++g) {
            const v2f b0 = *(const v2f*)(bx[g] + k);
            acc[g] = __builtin_amdgcn_wmma_f32_16x16x4_f32(
                false, a0, false, b0, (short)0, acc[g], false, false);
        }
#pragma unroll
        for (int g = 0; g < 4; ++g) {
            const v2f b1 = *(const v2f*)(bh[g] + k);
            acc[g] = __builtin_amdgcn_wmma_f32_16x16x4_f32(
                false, a1, false, b1, (short)0, acc[g], false, false);
        }
    }

    // K-split reduction through LDS (fixed summation order kq0+kq1+kq2+kq3)
    if (kq != 0) {
#pragma unroll
        for (int g = 0; g < 4; ++g)
#pragma unroll
            for (int r = 0; r < 8; ++r)
                red[tib][kq - 1][g][r][lane] = acc[g][r];
    }
    __syncthreads();
    if (kq != 0) return;

#pragma unroll
    for (int g = 0; g < 4; ++g)
#pragma unroll
        for (int r = 0; r < 8; ++r)
            acc[g][r] += red[tib][0][g][r][lane]
                       + red[tib][1][g][r][lane]
                       + red[tib][2][g][r][lane];

    // Fused LSTM cell epilogue. D layout: VGPR r holds M=r (lanes 0-15) / M=8+r (lanes 16-31).
#pragma unroll
    for (int r = 0; r < 8; ++r) {
        const int brow = m_tile * 16 + hiHalf * 8 + r;
        const size_t idx = (size_t)brow * HD + hcol;
        const float zi = acc[0][r];
        const float zf = acc[1][r];
        const float zg = acc[2][r];
        const float zo = acc[3][r];
        const float cold = cstate[idx];
        const float cn = sigf(zf) * cold + sigf(zi) * tanhf(zg);
        const float hn = sigf(zo) * tanhf(cn);
        cstate[idx] = cn;
        hout[idx]   = hn;
        if (seqout) seqout[(size_t)brow * (SD * HD) + hcol] = hn;
    }
}

// Tied-weight projection: out[m, v] = seq[m, :] . emb[v, :]
// M = B*S = 4096 (256 tiles), N = V = 32000 (2000 tiles), K = H = 1024.
// Each wave computes 4 M-tiles for one N-tile: one B (emb) load feeds 4 WMMAs,
// cutting L2 emb traffic 4x. 128000 waves = 16000 blocks x 256 threads.
// Output (524 MB) is stored non-temporally so it doesn't evict emb (131 MB)
// from the 192 MB L2 between M-tile passes.
__global__ __launch_bounds__(256)
void proj_kernel(const float* __restrict__ seq,
                 const float* __restrict__ emb,
                 float*       __restrict__ out)
{
    const int lane   = threadIdx.x & 31;
    const int wid    = (blockIdx.x * blockDim.x + threadIdx.x) >> 5; // 0..127999
    const int n_tile = wid % 2000;
    const int m4     = wid / 2000;        // 0..63  (4 M-tiles each)
    const int lane15 = lane & 15;
    const int hiHalf = (lane >> 4) & 1;
    const int koff   = hiHalf * 2;

    const float* ap[4];
#pragma unroll
    for (int i = 0; i < 4; ++i)
        ap[i] = seq + (size_t)(m4 * 64 + i * 16 + lane15) * HD + koff;
    const float* bp = emb + (size_t)(n_tile * 16 + lane15) * HD + koff;

    v8f acc[4];
#pragma unroll
    for (int i = 0; i < 4; ++i) acc[i] = (v8f){0, 0, 0, 0, 0, 0, 0, 0};

#pragma unroll 2
    for (int k = 0; k < HD; k += 4) {
        const v2f b = *(const v2f*)(bp + k);
#pragma unroll
        for (int i = 0; i < 4; ++i) {
            const v2f a = *(const v2f*)(ap[i] + k);
            acc[i] = __builtin_amdgcn_wmma_f32_16x16x4_f32(
                false, a, false, b, (short)0, acc[i], false, false);
        }
    }

    const int col = n_tile * 16 + lane15;
#pragma unroll
    for (int i = 0; i < 4; ++i) {
#pragma unroll
        for (int r = 0; r < 8; ++r) {
            const int row = m4 * 64 + i * 16 + hiHalf * 8 + r;
            __builtin_nontemporal_store(acc[i][r], out + (size_t)row * VD + col);
        }
    }
}

extern "C" void kernel_launch(void* const* d_in, const int* in_sizes, int n_in,
                              void* d_out, int out_size, void* d_ws, size_t ws_size,
                              hipStream_t stream) {
    // Input order: x, hidden, cell, target, tf_ratio, emb, w_ih, w_hh, b_ih, b_hh
    const int*   x      = (const int*)  d_in[0];   // [B,1]
    const float* hidden = (const float*)d_in[1];   // [L,B,H]
    const float* cell   = (const float*)d_in[2];   // [L,B,H]
    const int*   target = (const int*)  d_in[3];   // [B,S]
    const float* emb    = (const float*)d_in[5];   // [V,H]
    const float* w_ih   = (const float*)d_in[6];   // [L,4H,H]
    const float* w_hh   = (const float*)d_in[7];   // [L,4H,H]
    const float* b_ih   = (const float*)d_in[8];   // [L,4H]
    const float* b_hh   = (const float*)d_in[9];   // [L,4H]
    float* out = (float*)d_out;                    // [B,S,V]

    const int LBH = 2 * BD * HD;                   // 65536 floats per state set
    const int BH  = BD * HD;                       // 32768
    float* ws   = (float*)d_ws;
    float* c_st = ws;                              // [L,B,H], in-place updated
    float* h_a  = ws + LBH;                        // h double-buffer A
    float* h_b  = ws + 2 * LBH;                    // h double-buffer B
    float* seq  = ws + 3 * LBH;                    // [B,S,H] layer-1 outputs (~16 MB)

    hipMemcpyAsync(h_a,  hidden, (size_t)LBH * sizeof(float), hipMemcpyDeviceToDevice, stream);
    hipMemcpyAsync(c_st, cell,   (size_t)LBH * sizeof(float), hipMemcpyDeviceToDevice, stream);

    const size_t wOff = (size_t)4 * HD * HD;       // per-layer weight stride
    for (int t = 0; t < SD; ++t) {
        float* hc = (t & 1) ? h_b : h_a;           // h at step t-1
        float* hn = (t & 1) ? h_a : h_b;           // h at step t
        const int* tok       = (t == 0) ? x : (target + t);
        const int tok_stride = (t == 0) ? 1 : SD;

        // layer 0 (input = embedded token)
        lstm_step_kernel<<<64, 256, 0, stream>>>(
            emb, tok, tok_stride, nullptr,
            hc, c_st,
            w_ih, w_hh, b_ih, b_hh,
            hn, nullptr);
        // layer 1 (input = layer-0 h of this step), writes sequence output
        lstm_step_kernel<<<64, 256, 0, stream>>>(
            emb, nullptr, 0, hn,
            hc + BH, c_st + BH,
            w_ih + wOff, w_hh + wOff, b_ih + 4 * HD, b_hh + 4 * HD,
            hn + BH, seq + (size_t)t * HD);
    }

    // Final projection to vocabulary (tied embedding weights)
    proj_kernel<<<16000, 256, 0, stream>>>(seq, emb, out);
}